// SchNetUpdate_3874060501586
// MI455X (gfx1250) — compile-verified
//
#include <hip/hip_runtime.h>
#include <math.h>

// ---------------------------------------------------------------------------
// SchNet update for MI455X (gfx1250).
// Memory-bound at ~27 FLOP/byte (<< WMMA ridge), so keep fp32 end-to-end and
// run every GEMM through V_WMMA_F32_16X16X4_F32.
// B tiles stored TRANSPOSED in LDS so every WMMA fragment is one ds_load_b64
// (conflict-free with stride 132); K-loops fully unrolled so the scheduler can
// pipeline ds loads across WMMA groups; square A tiles staged to LDS with
// GLOBAL_LOAD_ASYNC_TO_LDS_B128 when available.
// ---------------------------------------------------------------------------

#define N_NODES 50000
#define N_EDGES 625000
#define H_NF    128
#define NFILT   128
#define N_GAUSS 50
#define LDSP    132           // A/B tile row stride (floats): 16B-aligned rows,
                              // banks advance by 4 per row -> conflict-free frags
#define W1S     56            // stride for transposed 52-row W_e1 tile
#define LN2     0.69314718055994531f

typedef float v2f __attribute__((ext_vector_type(2)));
typedef float v8f __attribute__((ext_vector_type(8)));
typedef int   v4i __attribute__((ext_vector_type(4)));

typedef __attribute__((address_space(1))) v4i* gv4i_p;   // global int4*
typedef __attribute__((address_space(3))) v4i* lv4i_p;   // LDS int4*

#if defined(__has_builtin)
# if __has_builtin(__builtin_amdgcn_global_load_async_to_lds_b128)
#  define USE_ASYNC_LDS 1
# endif
#endif
#ifndef USE_ASYNC_LDS
# define USE_ASYNC_LDS 0
#endif

__device__ __forceinline__ v8f wmma4(v2f a, v2f b, v8f c) {
  // D = A(16x4) * B(4x16) + C(16x16), fp32, wave32
  return __builtin_amdgcn_wmma_f32_16x16x4_f32(false, a, false, b, (short)0, c,
                                               false, false);
}

__device__ __forceinline__ float sspf(float x) {
  // shifted softplus, numerically stable
  return fmaxf(x, 0.f) + log1pf(__expf(-fabsf(x))) - LN2;
}

// One wave computes a 16x128 tile: rows [arow0, arow0+16) of Alds (row-major,
// stride LDSP) times B (transposed in LDS: BT[col*BSTRIDE + k]).
// A frag (ISA 7.12.2): lane l (0-15) holds A[M=l][K=k..k+1], lanes 16-31 hold
// A[M=l][K=k+2..k+3]; B frag mirrored; both are contiguous b64 LDS loads.
// C/D: vgpr r, half h -> row r+8h, col l.
template <int KDIM, int BSTRIDE>
__device__ __forceinline__ void wave_gemm(const float* Alds, int arow0,
                                          const float* BT, v8f acc[8]) {
  const int lane = threadIdx.x & 31;
  const int half = lane >> 4;
  const int l    = lane & 15;
  const float* arow = Alds + (arow0 + l) * LDSP + 2 * half;
  const float* bcol = BT + l * BSTRIDE + 2 * half;
#pragma unroll
  for (int k = 0; k < KDIM; k += 4) {
    v2f a = *(const v2f*)(arow + k);
#pragma unroll
    for (int n = 0; n < 8; ++n) {
      v2f b = *(const v2f*)(bcol + n * 16 * BSTRIDE + k);
      acc[n] = wmma4(a, b, acc[n]);
    }
  }
}

// cooperative transposed load of a 128x128 row-major matrix into LDS:
// dst[c*BSTRIDE + r] = src[r*128 + c]   (global reads coalesced in c)
template <int BSTRIDE>
__device__ __forceinline__ void load_w128_T(float* dst, const float* __restrict__ src) {
  for (int i = threadIdx.x; i < 128 * 128; i += blockDim.x) {
    int r = i >> 7, c = i & 127;
    dst[c * BSTRIDE + r] = src[i];
  }
}

// stage a 128-row x 128-col fp32 tile (src row stride = 128) into LDS rows of
// stride LDSP. Full in-range tiles go through the async LDS-DMA path.
__device__ __forceinline__ void stage_tile(float* dst, const float* src,
                                           int row0, int nrows_total) {
#if USE_ASYNC_LDS
  if (row0 + 128 <= nrows_total) {
    for (int g = (int)threadIdx.x; g < 128 * 32; g += (int)blockDim.x) {
      int r = g >> 5, c4 = (g & 31) << 2;   // 32 16-byte granules per row
      __builtin_amdgcn_global_load_async_to_lds_b128(
          (gv4i_p)(src + (size_t)(row0 + r) * 128 + c4),
          (lv4i_p)(dst + r * LDSP + c4),
          0, 0);
    }
    __asm__ volatile("s_wait_asynccnt 0x0" ::: "memory");
    return;
  }
#endif
  for (int i = (int)threadIdx.x; i < 128 * 128; i += (int)blockDim.x) {
    int r = i >> 7, c = i & 127;
    int gr = row0 + r;
    dst[r * LDSP + c] = (gr < nrows_total) ? src[(size_t)gr * 128 + c] : 0.f;
  }
}

// ---------------------------------------------------------------------------
// Kernel A: h = node_embs @ W_node_lin            [50000,128] x [128,128]
// ---------------------------------------------------------------------------
__global__ void __launch_bounds__(256) node_lin_kernel(
    const float* __restrict__ node_embs, const float* __restrict__ Wnl,
    float* __restrict__ h) {
  extern __shared__ float smem[];
  float* BwT = smem;                // 128 cols x LDSP
  float* At  = smem + 128 * LDSP;   // 128 rows x LDSP
  const int row0 = blockIdx.x * 128;

  load_w128_T<LDSP>(BwT, Wnl);
  stage_tile(At, node_embs, row0, N_NODES);
  __syncthreads();

  const int arow0 = (threadIdx.x >> 5) * 16;
  const int lane = threadIdx.x & 31;
  const int half = lane >> 4, l = lane & 15;

  v8f acc[8] = {};
  wave_gemm<128, LDSP>(At, arow0, BwT, acc);

#pragma unroll
  for (int r = 0; r < 8; ++r) {
    int gr = row0 + arow0 + r + 8 * half;
    if (gr < N_NODES) {
#pragma unroll
      for (int n = 0; n < 8; ++n)
        h[(size_t)gr * NFILT + n * 16 + l] = acc[n][r];
    }
  }
}

// ---------------------------------------------------------------------------
// zero the aggregation buffer (graph-capture safe; no memset)
// ---------------------------------------------------------------------------
__global__ void zero_kernel(float* __restrict__ p, int n) {
  int i = blockIdx.x * blockDim.x + threadIdx.x;
  int stride = gridDim.x * blockDim.x;
  for (; i < n; i += stride) p[i] = 0.f;
}

// ---------------------------------------------------------------------------
// Kernel B (hot): per 128-edge tile, fused
//   W = ssp(edge_embs @ W_e1 + b_e1) @ W_e2 + b_e2
//   edge_out = h[node_is] * (W * cutoff)
//   agg[node_js] += edge_out          (fp32 global atomics, L2-resident)
// ---------------------------------------------------------------------------
__global__ void __launch_bounds__(256) edge_kernel(
    const float* __restrict__ edge_embs, const float* __restrict__ edge_weights,
    const int* __restrict__ node_is, const int* __restrict__ node_js,
    const float* __restrict__ W_e1, const float* __restrict__ b_e1,
    const float* __restrict__ W_e2, const float* __restrict__ b_e2,
    const float* __restrict__ h, float* __restrict__ edge_out,
    float* __restrict__ agg) {
  extern __shared__ float smem[];
  float* W1T = smem;                        // 128 cols x W1S (k rows 50,51 = 0)
  float* W2T = smem + 128 * W1S;            // 128 cols x LDSP
  float* Et  = smem + 128 * (W1S + LDSP);   // 128 rows x LDSP: edge tile, then
                                            //   reused in place as activations
  const int e0 = blockIdx.x * 128;

  // W_e1 transposed, K padded 50 -> 52 with zeros
  for (int i = threadIdx.x; i < 52 * 128; i += blockDim.x) {
    int r = i >> 7, c = i & 127;
    W1T[c * W1S + r] = (r < N_GAUSS) ? W_e1[r * NFILT + c] : 0.f;
  }
  load_w128_T<LDSP>(W2T, W_e2);
  // edge feature tile (ragged 50-float rows -> guarded scalar staging)
  for (int i = threadIdx.x; i < 128 * 64; i += blockDim.x) {
    int r = i >> 6, c = i & 63;
    if (c < 52) {
      int e = e0 + r;
      Et[r * LDSP + c] =
          (e < N_EDGES && c < N_GAUSS) ? edge_embs[(size_t)e * N_GAUSS + c] : 0.f;
    }
  }
  __syncthreads();

  const int arow0 = (threadIdx.x >> 5) * 16;
  const int lane = threadIdx.x & 31;
  const int half = lane >> 4, l = lane & 15;

  // ---- layer 1: K = 52 (50 padded) ----
  v8f acc[8] = {};
  wave_gemm<52, W1S>(Et, arow0, W1T, acc);

  // SSP(x + b_e1) written back over this wave's own 16 rows of Et.
  // Per-wave LDS ops are in-order and no other wave touches these rows,
  // so no barrier is needed before layer 2.
#pragma unroll
  for (int n = 0; n < 8; ++n) {
    float b = b_e1[n * 16 + l];
#pragma unroll
    for (int r = 0; r < 8; ++r)
      Et[(arow0 + r + 8 * half) * LDSP + n * 16 + l] = sspf(acc[n][r] + b);
  }

  // ---- layer 2: K = 128 ----
  v8f acc2[8] = {};
  wave_gemm<128, LDSP>(Et, arow0, W2T, acc2);

  float be2[8];
#pragma unroll
  for (int n = 0; n < 8; ++n) be2[n] = b_e2[n * 16 + l];

  // ---- fused epilogue: cutoff * gather-multiply, store, scatter-add ----
#pragma unroll
  for (int r = 0; r < 8; ++r) {
    int e = e0 + arow0 + r + 8 * half;
    if (e < N_EDGES) {
      float w = edge_weights[e];
      // 0.5*(cos(w * PI/0.5)+1) * (w < 0.5)
      float cut = (w < 0.5f) ? 0.5f * (__cosf(w * 6.283185307179586f) + 1.f) : 0.f;
      int ni = node_is[e];
      int nj = node_js[e];
      const float* hrow = &h[(size_t)ni * NFILT];
      float* aggrow = &agg[(size_t)nj * NFILT];
      float* orow = &edge_out[(size_t)e * NFILT];
#pragma unroll
      for (int n = 0; n < 8; ++n) {
        int col = n * 16 + l;
        float val = hrow[col] * ((acc2[n][r] + be2[n]) * cut);
        orow[col] = val;
        __hip_atomic_fetch_add(&aggrow[col], val, __ATOMIC_RELAXED,
                               __HIP_MEMORY_SCOPE_AGENT);
      }
    }
  }
}

// ---------------------------------------------------------------------------
// Kernel C: node_out = ssp(agg @ W_n1 + b_n1) @ W_n2 + b_n2 ; out += residual
// ---------------------------------------------------------------------------
__global__ void __launch_bounds__(256) node_mlp_kernel(
    const float* __restrict__ agg, const float* __restrict__ node_embs,
    const float* __restrict__ W_n1, const float* __restrict__ b_n1,
    const float* __restrict__ W_n2, const float* __restrict__ b_n2,
    float* __restrict__ out_nodes) {
  extern __shared__ float smem[];
  float* W1T = smem;                 // 128 x LDSP
  float* W2T = smem + 128 * LDSP;    // 128 x LDSP
  float* At  = smem + 256 * LDSP;    // 128 x LDSP : agg tile, reused as act
  const int row0 = blockIdx.x * 128;

  load_w128_T<LDSP>(W1T, W_n1);
  load_w128_T<LDSP>(W2T, W_n2);
  stage_tile(At, agg, row0, N_NODES);
  __syncthreads();

  const int arow0 = (threadIdx.x >> 5) * 16;
  const int lane = threadIdx.x & 31;
  const int half = lane >> 4, l = lane & 15;

  v8f acc[8] = {};
  wave_gemm<128, LDSP>(At, arow0, W1T, acc);

#pragma unroll
  for (int n = 0; n < 8; ++n) {
    float b = b_n1[n * 16 + l];
#pragma unroll
    for (int r = 0; r < 8; ++r)
      At[(arow0 + r + 8 * half) * LDSP + n * 16 + l] = sspf(acc[n][r] + b);
  }

  v8f acc2[8] = {};
  wave_gemm<128, LDSP>(At, arow0, W2T, acc2);

#pragma unroll
  for (int r = 0; r < 8; ++r) {
    int gr = row0 + arow0 + r + 8 * half;
    if (gr < N_NODES) {
#pragma unroll
      for (int n = 0; n < 8; ++n) {
        int col = n * 16 + l;
        out_nodes[(size_t)gr * H_NF + col] =
            node_embs[(size_t)gr * H_NF + col] + acc2[n][r] + b_n2[col];
      }
    }
  }
}

// ---------------------------------------------------------------------------
extern "C" void kernel_launch(void* const* d_in, const int* in_sizes, int n_in,
                              void* d_out, int out_size, void* d_ws,
                              size_t ws_size, hipStream_t stream) {
  const float* node_embs    = (const float*)d_in[0];
  const float* edge_embs    = (const float*)d_in[1];
  const float* edge_weights = (const float*)d_in[2];
  const int*   edge_index   = (const int*)d_in[3];
  const float* W_node_lin   = (const float*)d_in[4];
  const float* W_e1 = (const float*)d_in[5];
  const float* b_e1 = (const float*)d_in[6];
  const float* W_e2 = (const float*)d_in[7];
  const float* b_e2 = (const float*)d_in[8];
  const float* W_n1 = (const float*)d_in[9];
  const float* b_n1 = (const float*)d_in[10];
  const float* W_n2 = (const float*)d_in[11];
  const float* b_n2 = (const float*)d_in[12];

  float* out_nodes = (float*)d_out;                             // [N, h_nf]
  float* out_edges = (float*)d_out + (size_t)N_NODES * H_NF;    // [E, F]

  float* h   = (float*)d_ws;                                    // [N, F]
  float* agg = (float*)d_ws + (size_t)N_NODES * NFILT;          // [N, F]

  const int* node_is = edge_index;            // edge_index[0]
  const int* node_js = edge_index + N_EDGES;  // edge_index[1]

  const size_t lds_A = (size_t)(2 * 128) * LDSP * sizeof(float);          // 132 KB
  const size_t lds_B = (size_t)128 * (W1S + 2 * LDSP) * sizeof(float);    // 160 KB
  const size_t lds_C = (size_t)(3 * 128) * LDSP * sizeof(float);          // 198 KB

  node_lin_kernel<<<(N_NODES + 127) / 128, 256, lds_A, stream>>>(
      node_embs, W_node_lin, h);

  zero_kernel<<<1024, 256, 0, stream>>>(agg, N_NODES * NFILT);

  edge_kernel<<<(N_EDGES + 127) / 128, 256, lds_B, stream>>>(
      edge_embs, edge_weights, node_is, node_js, W_e1, b_e1, W_e2, b_e2, h,
      out_edges, agg);

  node_mlp_kernel<<<(N_NODES + 127) / 128, 256, lds_C, stream>>>(
      agg, node_embs, W_n1, b_n1, W_n2, b_n2, out_nodes);
}